// Model_1640677507285
// MI455X (gfx1250) — compile-verified
//
#include <hip/hip_runtime.h>
#include <math.h>

typedef __attribute__((ext_vector_type(2))) float v2f;
typedef __attribute__((ext_vector_type(8))) float v8f;

#define B_ 16
#define N_ 512
#define H_ 8
#define F_ 128
#define NEG_SLOPE_ 0.2f

// ---------------------------------------------------------------------------
// P1: fold attention vectors into W:  wa_src[h][i] = sum_f W[h,i,f]*a_src[h,f]
// (linearity: s_src[b,h,n] = h[b,n,:] . wa_src[h,:])
// ---------------------------------------------------------------------------
__global__ void gat_prep_wa(const float* __restrict__ W,
                            const float* __restrict__ a,
                            float* __restrict__ wa_src,
                            float* __restrict__ wa_dst) {
  const int hh = blockIdx.x;
  const int i  = threadIdx.x;
  const float* wrow = W + ((size_t)hh * F_ + i) * F_;
  const float* as   = a + (size_t)hh * 2 * F_;
  const float* ad   = as + F_;
  float s0 = 0.f, s1 = 0.f;
  #pragma unroll 4
  for (int f = 0; f < F_; ++f) {
    const float w = wrow[f];
    s0 += w * as[f];
    s1 += w * ad[f];
  }
  wa_src[hh * F_ + i] = s0;
  wa_dst[hh * F_ + i] = s1;
}

// ---------------------------------------------------------------------------
// P2: Wt[h][f][i] = W[h][i][f]  (so K1 B-fragments are contiguous b64 loads)
// ---------------------------------------------------------------------------
__global__ void gat_transpose_w(const float* __restrict__ W,
                                float* __restrict__ Wt) {
  const int idx = blockIdx.x * 256 + threadIdx.x;   // H*F*F total
  const int hh  = idx >> 14;
  const int rem = idx & 16383;
  const int f   = rem >> 7;
  const int i   = rem & 127;
  Wt[((size_t)hh * F_ + f) * F_ + i] = W[((size_t)hh * F_ + i) * F_ + f];
}

// ---------------------------------------------------------------------------
// K2: s_src[b,h,n] = h[b,n,:] . wa_src[h,:]  (one wave per row, wave32 shfl)
// ---------------------------------------------------------------------------
__launch_bounds__(256)
__global__ void gat_scores_kernel(const float* __restrict__ h,
                                  const float* __restrict__ wa_src,
                                  const float* __restrict__ wa_dst,
                                  float* __restrict__ s_src,
                                  float* __restrict__ s_dst) {
  const int row  = blockIdx.x * (blockDim.x >> 5) + (threadIdx.x >> 5);
  const int lane = threadIdx.x & 31;
  const int b    = row / (H_ * N_);
  const int hh   = (row / N_) & (H_ - 1);
  const int n    = row & (N_ - 1);

  const float4 x  = ((const float4*)(h + ((size_t)b * N_ + n) * F_))[lane];
  const float4 ws = ((const float4*)(wa_src + (size_t)hh * F_))[lane];
  const float4 wd = ((const float4*)(wa_dst + (size_t)hh * F_))[lane];

  float ss = x.x * ws.x + x.y * ws.y + x.z * ws.z + x.w * ws.w;
  float sd = x.x * wd.x + x.y * wd.y + x.z * wd.z + x.w * wd.w;
  #pragma unroll
  for (int off = 16; off >= 1; off >>= 1) {
    ss += __shfl_xor(ss, off, 32);
    sd += __shfl_xor(sd, off, 32);
  }
  if (lane == 0) {
    s_src[row] = ss;
    s_dst[row] = sd;
  }
}

// ---------------------------------------------------------------------------
// K1: hpT[b,h,f,n] = sum_i h[b,n,i] * W[h,i,f]   (TRANSPOSED store)
// One wave = one 16(n) x 16(f) tile via V_WMMA_F32_16X16X4_F32.
// A loads: b64 (h rows). B loads: b64 (Wt rows). Stores: 2x b128 per lane.
// ---------------------------------------------------------------------------
__launch_bounds__(256)
__global__ void gat_hprime_kernel(const float* __restrict__ h,
                                  const float* __restrict__ Wt,
                                  float* __restrict__ hpT) {
  const int rt   = blockIdx.x;          // row tile over flattened B*N
  const int hh   = blockIdx.y;          // head
  const int wave = threadIdx.x >> 5;
  const int lane = threadIdx.x & 31;
  const int lm   = lane & 15;
  const int lh   = lane >> 4;
  const int f0   = wave * 16;
  const int row0 = rt * 16;

  // A: lane l, vgpr v: M = l%16, K = (l/16)*2 + v
  const float* hrow = h + (size_t)(row0 + lm) * F_;
  // B: lane l, vgpr v: N(f) = l%16, K = (l/16)*2 + v  -> Wt[f0+lm][k..]
  const float* wtb  = Wt + ((size_t)hh * F_ + f0 + lm) * F_;

  v8f c = {0.f, 0.f, 0.f, 0.f, 0.f, 0.f, 0.f, 0.f};
  #pragma unroll 4
  for (int k = 0; k < F_; k += 4) {
    const v2f a  = *(const v2f*)(hrow + k + lh * 2);
    const v2f bb = *(const v2f*)(wtb  + k + lh * 2);
    c = __builtin_amdgcn_wmma_f32_16x16x4_f32(false, a, false, bb,
                                              (short)0, c, false, false);
  }

  // C: vgpr r -> row M = r + lh*8, col f = f0+lm. Transposed store:
  // hpT[((b*H+hh)*F + f)*N + n], c[0..7] map to 8 consecutive n.
  const int b  = row0 >> 9;
  const int n0 = row0 & (N_ - 1);
  float* obase = hpT + (((size_t)b * H_ + hh) * F_ + f0 + lm) * N_ + n0 + lh * 8;
  *(float4*)(obase)     = make_float4(c[0], c[1], c[2], c[3]);
  *(float4*)(obase + 4) = make_float4(c[4], c[5], c[6], c[7]);
}

// ---------------------------------------------------------------------------
// K3: softmax + aggregation + head-mean.
// One workgroup per (b, 16-row i-tile); 8 waves = 8 heads.
// Pass 1: row max (b64 score loads). Pass 2: WMMA over j (K=4 chunks),
// accumulating UNNORMALIZED exp attention + its row sum; C rescaled by 1/sum
// at the end (shfl broadcast). B fragments are b64 loads from hpT.
// Cross-head mean via 64KB LDS reduction (deterministic).
// ---------------------------------------------------------------------------
__device__ __forceinline__ float gat_score(float ssrc, float sdst,
                                           float bias, int adjv) {
  float s = ssrc + sdst;
  s = (s > 0.f) ? s : NEG_SLOPE_ * s;
  s += bias;
  return (adjv != 0) ? s : -__builtin_inff();
}

__launch_bounds__(256)
__global__ void gat_attn_kernel(const float* __restrict__ hpT,
                                const float* __restrict__ s_src,
                                const float* __restrict__ s_dst,
                                const float* __restrict__ bias,
                                const int* __restrict__ adj,
                                float* __restrict__ out) {
  __shared__ float red[H_ * 16 * F_];   // 64 KB

  const int it   = blockIdx.x;          // i-tile
  const int b    = blockIdx.y;          // batch
  const int hh   = threadIdx.x >> 5;    // wave == head
  const int lane = threadIdx.x & 31;
  const int lm   = lane & 15;
  const int lh   = lane >> 4;
  const int n0   = it * 16;

  const size_t bh   = (size_t)b * H_ + hh;
  const float  ssrc = s_src[bh * N_ + n0 + lm];     // row i = n0+lm
  const float* sdr  = s_dst + bh * N_;
  const float* br   = bias + (size_t)(n0 + lm) * N_;
  const int*   ar   = adj + (size_t)(n0 + lm) * N_;
  const float* bpb  = hpT + bh * F_ * N_;           // [f][j] panel

  // Prefetch the first cachelines of the 8 f-rows this lane will stream.
  #pragma unroll
  for (int t = 0; t < 8; ++t)
    __builtin_prefetch(bpb + (size_t)(t * 16 + lm) * N_, 0, 0);

  // pass 1: row max. lane (lm,lh) covers j with j%4 in {2lh, 2lh+1}.
  float m = -__builtin_inff();
  for (int t = 0; t < N_ / 4; ++t) {
    const int j = 4 * t + lh * 2;
    const v2f  sd = *(const v2f*)(sdr + j);
    const v2f  bv = *(const v2f*)(br + j);
    const int2 av = *(const int2*)(ar + j);
    m = fmaxf(m, gat_score(ssrc, sd.x, bv.x, av.x));
    m = fmaxf(m, gat_score(ssrc, sd.y, bv.y, av.y));
  }
  m = fmaxf(m, __shfl_xor(m, 16, 32));  // both half-waves hold row max

  // pass 2: accumulate exp(s-m) * hp with WMMA; track row sum on the fly.
  v8f c[8];
  #pragma unroll
  for (int t = 0; t < 8; ++t)
    c[t] = (v8f){0.f, 0.f, 0.f, 0.f, 0.f, 0.f, 0.f, 0.f};
  float sum = 0.f;

  for (int jc = 0; jc < N_; jc += 4) {
    const int j0 = jc + lh * 2;         // A frag: M=lm, K=lh*2+v
    const v2f  sd = *(const v2f*)(sdr + j0);
    const v2f  bv = *(const v2f*)(br + j0);
    const int2 av = *(const int2*)(ar + j0);
    v2f afrag;
    afrag.x = __expf(gat_score(ssrc, sd.x, bv.x, av.x) - m);
    afrag.y = __expf(gat_score(ssrc, sd.y, bv.y, av.y) - m);
    sum += afrag.x + afrag.y;

    #pragma unroll
    for (int t = 0; t < 8; ++t) {
      // B frag: N(f)=lm, K=lh*2+v -> hpT row f0+lm, consecutive j -> b64
      const v2f bfrag = *(const v2f*)(bpb + (size_t)(t * 16 + lm) * N_ + j0);
      c[t] = __builtin_amdgcn_wmma_f32_16x16x4_f32(false, afrag, false, bfrag,
                                                   (short)0, c[t], false, false);
    }
  }
  sum += __shfl_xor(sum, 16, 32);
  const float inv = 1.f / sum;

  // Rescale C rows: row of vgpr r is (r + lh*8); its 1/sum lives in lane (r+lh*8).
  #pragma unroll
  for (int r = 0; r < 8; ++r) {
    const float invr = __shfl(inv, r + (lh << 3), 32);
    #pragma unroll
    for (int t = 0; t < 8; ++t) c[t][r] *= invr;
  }

  // spill per-head tiles to LDS, reduce across heads (deterministic)
  float* myred = red + hh * (16 * F_);
  #pragma unroll
  for (int t = 0; t < 8; ++t) {
    #pragma unroll
    for (int r = 0; r < 8; ++r) {
      const int i = r + lh * 8;
      myred[i * F_ + t * 16 + lm] = c[t][r];
    }
  }
  __syncthreads();

  for (int idx = threadIdx.x; idx < 16 * F_; idx += 256) {
    float acc = 0.f;
    #pragma unroll
    for (int w = 0; w < H_; ++w) acc += red[w * 16 * F_ + idx];
    const int i = idx >> 7;
    const int f = idx & (F_ - 1);
    out[((size_t)b * N_ + n0 + i) * F_ + f] = acc * (1.0f / H_);
  }
}

// ---------------------------------------------------------------------------
extern "C" void kernel_launch(void* const* d_in, const int* in_sizes, int n_in,
                              void* d_out, int out_size, void* d_ws, size_t ws_size,
                              hipStream_t stream) {
  const float* h    = (const float*)d_in[0];  // (B, N, F_IN) f32
  const int*   adj  = (const int*)d_in[1];    // (N, N) i32
  const float* bias = (const float*)d_in[2];  // (1, 1, N, N) f32
  const float* W    = (const float*)d_in[3];  // (H, F_IN, F_OUT) f32
  const float* a    = (const float*)d_in[4];  // (H, 2*F_OUT, 1) f32
  float* out = (float*)d_out;                 // (B, N, F_OUT) f32

  float* hpT    = (float*)d_ws;                          // B*H*F*N floats (32 MB)
  float* s_src  = hpT + (size_t)B_ * H_ * F_ * N_;       // B*H*N
  float* s_dst  = s_src + (size_t)B_ * H_ * N_;          // B*H*N
  float* wa_src = s_dst + (size_t)B_ * H_ * N_;          // H*F
  float* wa_dst = wa_src + (size_t)H_ * F_;              // H*F
  float* Wt     = wa_dst + (size_t)H_ * F_;              // H*F*F (512 KB)

  gat_prep_wa<<<H_, 128, 0, stream>>>(W, a, wa_src, wa_dst);
  gat_transpose_w<<<(H_ * F_ * F_) / 256, 256, 0, stream>>>(W, Wt);

  const int rows = B_ * H_ * N_;
  gat_scores_kernel<<<rows / 8, 256, 0, stream>>>(h, wa_src, wa_dst, s_src, s_dst);

  dim3 g1((B_ * N_) / 16, H_);
  gat_hprime_kernel<<<g1, 256, 0, stream>>>(h, Wt, hpT);

  dim3 g3(N_ / 16, B_);
  gat_attn_kernel<<<g3, 256, 0, stream>>>(hpT, s_src, s_dst, bias, adj, out);
}